// SimplePathHelper_41188736369266
// MI455X (gfx1250) — compile-verified
//
#include <hip/hip_runtime.h>

#define N_SEGC  10000
#define THREADS 256
#define BLOCKS  2048

typedef __attribute__((address_space(3))) float lds_f32;

__global__ __launch_bounds__(THREADS) void path_eval_kernel(
    const float* __restrict__ xstart,
    const float* __restrict__ dx,
    const float* __restrict__ cps,   // (N_SEGC, 4, 3) row-major
    const float* __restrict__ s,
    float*       __restrict__ out_pos,
    int*         __restrict__ out_idx,
    int M)
{
    __shared__ float sx[N_SEGC];            // 40,000 bytes
    lds_f32* lsx = (lds_f32*)sx;

    // ---- Stage xstart into LDS via CDNA5 async global->LDS DMA path ----
    // 10000 floats = 2500 x 16B chunks, 16B aligned.
    for (int c = threadIdx.x; c < N_SEGC / 4; c += THREADS) {
        unsigned lds_addr = (unsigned)(unsigned long long)(lsx + 4 * c);
        const float* gaddr = xstart + 4 * c;
        asm volatile("global_load_async_to_lds_b128 %0, %1, off"
                     :: "v"(lds_addr), "v"(gaddr)
                     : "memory");
    }
#if __has_builtin(__builtin_amdgcn_s_wait_asynccnt)
    __builtin_amdgcn_s_wait_asynccnt(0);
#else
    asm volatile("s_wait_asynccnt 0" ::: "memory");
#endif
    __syncthreads();

    const float xlast = sx[N_SEGC - 1];
    const float total = xlast + dx[N_SEGC - 1];   // uniform scalar load

    const int stride = gridDim.x * blockDim.x;
    for (int m = blockIdx.x * blockDim.x + threadIdx.x; m < M; m += stride) {
        const float sv = s[m];

        // x = mod(s, total), clamped into [0, total]
        float x = sv - floorf(sv / total) * total;
        x = fminf(fmaxf(x, 0.0f), total);

        // Branch-free binary search in LDS:
        // pos = #elements of xstart <= x  (searchsorted(..., 'right'))
        int pos = 0;
#pragma unroll
        for (int step = 8192; step > 0; step >>= 1) {
            const int np = pos + step;
            const int a  = (np <= N_SEGC) ? (np - 1) : (N_SEGC - 1);
            const bool le = (sx[a] <= x) && (np <= N_SEGC);
            pos = le ? np : pos;
        }
        int idx = pos - 1;
        idx = max(0, min(idx, N_SEGC - 1));

        // dx[idx] reconstructed from LDS (xstart is cumsum of dx)
        const float xs = sx[idx];
        const float xe = (idx < N_SEGC - 1) ? sx[idx + 1] : total;
        const float t  = (x - xs) / (xe - xs);
        const float u  = 1.0f - t;

        const float b0 = u * u * u;
        const float b1 = 3.0f * t * (u * u);
        const float b2 = 3.0f * (t * t) * u;
        const float b3 = t * (t * t);

        // 12 floats per segment = three aligned b128 loads
        const float4* cp = (const float4*)(cps + (size_t)idx * 12);
        const float4 c0 = cp[0];   // p0x p0y p0z p1x
        const float4 c1 = cp[1];   // p1y p1z p2x p2y
        const float4 c2 = cp[2];   // p2z p3x p3y p3z

        const float px = b0 * c0.x + b1 * c0.w + b2 * c1.z + b3 * c2.y;
        const float py = b0 * c0.y + b1 * c1.x + b2 * c1.w + b3 * c2.z;
        const float pz = b0 * c0.z + b1 * c1.y + b2 * c2.x + b3 * c2.w;

        out_pos[3 * m + 0] = px;
        out_pos[3 * m + 1] = py;
        out_pos[3 * m + 2] = pz;
        out_idx[m] = idx;
    }
}

extern "C" void kernel_launch(void* const* d_in, const int* in_sizes, int n_in,
                              void* d_out, int out_size, void* d_ws, size_t ws_size,
                              hipStream_t stream) {
    (void)n_in; (void)out_size; (void)d_ws; (void)ws_size;
    const float* xstart = (const float*)d_in[0];   // (10000,)
    const float* dx     = (const float*)d_in[1];   // (10000,)
    const float* cps    = (const float*)d_in[2];   // (10000, 4, 3)
    const float* s      = (const float*)d_in[3];   // (M,)
    const int M = in_sizes[3];

    float* out_pos = (float*)d_out;                 // M*3 floats
    int*   out_idx = (int*)d_out + (size_t)3 * M;   // M int32 (bit-cast region)

    path_eval_kernel<<<BLOCKS, THREADS, 0, stream>>>(
        xstart, dx, cps, s, out_pos, out_idx, M);
}